// VectorQuantizer_31602369364525
// MI455X (gfx1250) — compile-verified
//
#include <hip/hip_runtime.h>

// ---------------------------------------------------------------------------
// VQ-VAE vector quantizer for MI455X (gfx1250, wave32, WMMA)
//   z:            [131072, 64] f32
//   codebook:     [1024, 64]   f32
//   cluster_size: [1024]       f32
//   ema_w:        [1024, 64]   f32
// Outputs (flat f32, in reference return order):
//   quantized_st [N*D], indices [N] (as float), loss [1],
//   new_codebook [K*D], new_cluster_size [K], new_ema_w [K*D]
//
// v2: two 16-token tiles per wave -> each codebook B fragment feeds two
//     independent v_wmma_f32_16x16x4_f32 chains (ILP + half the DS traffic),
//     grid halved to 512 blocks (half the codebook->LDS staging).
// ---------------------------------------------------------------------------

#define NUM_CODES   1024
#define CODE_DIM    64
#define N_TOKENS    131072
#define DECAY       0.99f
#define OMD         0.01f     // 1 - DECAY
#define EPSV        1e-5f

#define CB_STRIDE   68        // padded LDS row stride (floats): avoids bank conflicts
#define TOK_PER_WAVE 32
#define TOK_PER_BLOCK 256     // 8 waves * 32 tokens

typedef __attribute__((ext_vector_type(2))) float v2f;
typedef __attribute__((ext_vector_type(8))) float v8f;

// workspace layout (floats): [0,1024) counts | [1024, 1024+65536) dw | [66560] loss
#define WS_COUNTS  0
#define WS_DW      1024
#define WS_LOSS    (1024 + 65536)
#define WS_FLOATS  (1024 + 65536 + 1)

__global__ void vq_zero_kernel(float* __restrict__ ws, int n) {
    int i = blockIdx.x * blockDim.x + threadIdx.x;
    if (i < n) ws[i] = 0.0f;
}

__global__ __launch_bounds__(256)
void vq_main_kernel(const float* __restrict__ z,
                    const float* __restrict__ cbg,
                    float* __restrict__ out_q,
                    float* __restrict__ out_idx,
                    float* __restrict__ ws_counts,
                    float* __restrict__ ws_dw,
                    float* __restrict__ ws_loss)
{
    __shared__ float cb[NUM_CODES * CB_STRIDE];  // codebook, padded rows (~278 KB)
    __shared__ float cnorm[NUM_CODES];           // ||c||^2
    __shared__ int   widx[TOK_PER_BLOCK];        // winning index per token

    const int tid = threadIdx.x;

    // ---- Stage 1: whole codebook -> LDS (vectorized, coalesced) ----
    for (int i = tid; i < NUM_CODES * (CODE_DIM / 4); i += 256) {
        const int   row = i >> 4;           // / (64/4)
        const int   c4  = (i & 15) << 2;
        const float4 v  = ((const float4*)cbg)[i];
        float* dst = &cb[row * CB_STRIDE + c4];
        dst[0] = v.x; dst[1] = v.y; dst[2] = v.z; dst[3] = v.w;
    }
    __syncthreads();

    for (int c = tid; c < NUM_CODES; c += 256) {
        const float* r = &cb[c * CB_STRIDE];
        float s = 0.0f;
        #pragma unroll
        for (int d = 0; d < CODE_DIM; ++d) s += r[d] * r[d];
        cnorm[c] = s;
    }
    __syncthreads();

    // ---- Stage 2: per-wave 2x16-token tiles vs all 1024 codes via WMMA f32 ----
    const int lane = tid & 31;
    const int wave = tid >> 5;
    const int half = lane >> 4;     // selects K-pair {0,1} vs {2,3}
    const int lrow = lane & 15;     // matrix row (A) / column (B,C,D)
    const int tokBase = blockIdx.x * TOK_PER_BLOCK + wave * TOK_PER_WAVE;

    // A fragments for 16 K-steps, two token tiles
    v2f afrag0[16], afrag1[16];
    {
        const float2* zr0 = (const float2*)(z + (size_t)(tokBase + lrow) * CODE_DIM);
        const float2* zr1 = (const float2*)(z + (size_t)(tokBase + 16 + lrow) * CODE_DIM);
        #pragma unroll
        for (int s = 0; s < 16; ++s) {
            float2 v0 = zr0[2 * s + half];
            float2 v1 = zr1[2 * s + half];
            afrag0[s].x = v0.x; afrag0[s].y = v0.y;
            afrag1[s].x = v1.x; afrag1[s].y = v1.y;
        }
    }

    float bestv0[8], bestv1[8];
    int   bidx0[8],  bidx1[8];
    #pragma unroll
    for (int r = 0; r < 8; ++r) {
        bestv0[r] = 3.4e38f; bidx0[r] = 0;
        bestv1[r] = 3.4e38f; bidx1[r] = 0;
    }

    for (int t = 0; t < NUM_CODES / 16; ++t) {
        v8f acc0 = {};   // two independent accumulator chains
        v8f acc1 = {};
        const float* brow = &cb[(t * 16 + lrow) * CB_STRIDE + 2 * half];
        #pragma unroll
        for (int s = 0; s < 16; ++s) {
            v2f b;
            b.x = brow[4 * s];
            b.y = brow[4 * s + 1];
            acc0 = __builtin_amdgcn_wmma_f32_16x16x4_f32(
                false, afrag0[s], false, b, (short)0, acc0, false, false);
            acc1 = __builtin_amdgcn_wmma_f32_16x16x4_f32(
                false, afrag1[s], false, b, (short)0, acc1, false, false);
        }
        // lane covers code n = t*16+lrow; rows m = r + 8*half per tile
        const float cn = cnorm[t * 16 + lrow];
        const int   gi = t * 16 + lrow;
        #pragma unroll
        for (int r = 0; r < 8; ++r) {
            const float s0 = cn - 2.0f * acc0[r];
            const float s1 = cn - 2.0f * acc1[r];
            if (s0 < bestv0[r]) { bestv0[r] = s0; bidx0[r] = gi; }
            if (s1 < bestv1[r]) { bestv1[r] = s1; bidx1[r] = gi; }
        }
    }

    // Cross-lane argmin within each 16-lane half (token rows fixed per half)
    #pragma unroll
    for (int r = 0; r < 8; ++r) {
        float v0 = bestv0[r]; int ix0 = bidx0[r];
        float v1 = bestv1[r]; int ix1 = bidx1[r];
        #pragma unroll
        for (int m = 8; m >= 1; m >>= 1) {
            const float ov0 = __shfl_xor(v0, m, 32);
            const int   oi0 = __shfl_xor(ix0, m, 32);
            const float ov1 = __shfl_xor(v1, m, 32);
            const int   oi1 = __shfl_xor(ix1, m, 32);
            if (ov0 < v0 || (ov0 == v0 && oi0 < ix0)) { v0 = ov0; ix0 = oi0; }
            if (ov1 < v1 || (ov1 == v1 && oi1 < ix1)) { v1 = ov1; ix1 = oi1; }
        }
        if (lrow == 0) {
            widx[wave * TOK_PER_WAVE + half * 8 + r]      = ix0;  // tile 0: token m
            widx[wave * TOK_PER_WAVE + 16 + half * 8 + r] = ix1;  // tile 1: token 16+m
        }
    }
    __syncthreads();

    // ---- Stage 3: gather quantized rows, loss partials, segment sums ----
    float lossAcc = 0.0f;
    const int d0 = 2 * lane;  // each lane owns 2 of the 64 dims
    for (int tk = 0; tk < TOK_PER_WAVE; ++tk) {
        const int idx = widx[wave * TOK_PER_WAVE + tk];
        const int tok = tokBase + tk;
        const float q0 = cb[idx * CB_STRIDE + d0];
        const float q1 = cb[idx * CB_STRIDE + d0 + 1];
        const float z0 = z[(size_t)tok * CODE_DIM + d0];
        const float z1 = z[(size_t)tok * CODE_DIM + d0 + 1];
        float2 o; o.x = q0; o.y = q1;
        *(float2*)&out_q[(size_t)tok * CODE_DIM + d0] = o;     // quantized_st value == q
        lossAcc += (z0 - q0) * (z0 - q0) + (z1 - q1) * (z1 - q1);
        atomicAdd(&ws_dw[idx * CODE_DIM + d0],     z0);        // dw = segment_sum(z)
        atomicAdd(&ws_dw[idx * CODE_DIM + d0 + 1], z1);
        if (lane == 0) atomicAdd(&ws_counts[idx], 1.0f);       // counts
    }
    out_idx[tokBase + lane] = (float)widx[wave * TOK_PER_WAVE + lane];

    #pragma unroll
    for (int m = 16; m >= 1; m >>= 1) lossAcc += __shfl_xor(lossAcc, m, 32);
    if (lane == 0) atomicAdd(ws_loss, lossAcc);
}

__global__ __launch_bounds__(1024)
void vq_finalize_kernel(const float* __restrict__ cluster_size,
                        const float* __restrict__ ema_w,
                        const float* __restrict__ ws_counts,
                        const float* __restrict__ ws_dw,
                        const float* __restrict__ ws_loss,
                        float* __restrict__ out_loss,
                        float* __restrict__ out_cb,
                        float* __restrict__ out_cs,
                        float* __restrict__ out_ema)
{
    __shared__ float red[NUM_CODES];
    __shared__ float csArr[NUM_CODES];
    const int k = threadIdx.x;

    const float ncs = DECAY * cluster_size[k] + OMD * ws_counts[k];
    out_cs[k] = ncs;
    red[k] = ncs;
    __syncthreads();
    for (int off = 512; off >= 1; off >>= 1) {
        if (k < off) red[k] += red[k + off];
        __syncthreads();
    }
    const float n = red[0];
    csArr[k] = (ncs + EPSV) / (n + NUM_CODES * EPSV) * n;  // Laplace-smoothed size
    __syncthreads();

    for (int j = 0; j < CODE_DIM; ++j) {
        const int i   = j * NUM_CODES + k;   // coalesced over K*D
        const int row = i >> 6;              // / CODE_DIM
        const float nw = DECAY * ema_w[i] + OMD * ws_dw[i];
        out_ema[i] = nw;
        out_cb[i]  = nw / csArr[row];
    }
    if (k == 0) out_loss[0] = ws_loss[0] / (float)((size_t)N_TOKENS * CODE_DIM);
}

extern "C" void kernel_launch(void* const* d_in, const int* in_sizes, int n_in,
                              void* d_out, int out_size, void* d_ws, size_t ws_size,
                              hipStream_t stream) {
    (void)in_sizes; (void)n_in; (void)out_size; (void)ws_size;
    const float* z            = (const float*)d_in[0];
    const float* codebook     = (const float*)d_in[1];
    const float* cluster_size = (const float*)d_in[2];
    const float* ema_w        = (const float*)d_in[3];

    float* out = (float*)d_out;
    float* ws  = (float*)d_ws;

    float* ws_counts = ws + WS_COUNTS;
    float* ws_dw     = ws + WS_DW;
    float* ws_loss   = ws + WS_LOSS;

    // output offsets (floats), reference return order
    const size_t OQ  = 0;                                  // quantized_st  [N*D]
    const size_t OI  = OQ  + (size_t)N_TOKENS * CODE_DIM;  // indices       [N]
    const size_t OL  = OI  + N_TOKENS;                     // loss          [1]
    const size_t OCB = OL  + 1;                            // new_codebook  [K*D]
    const size_t OCS = OCB + (size_t)NUM_CODES * CODE_DIM; // new_cluster   [K]
    const size_t OEW = OCS + NUM_CODES;                    // new_ema_w     [K*D]

    vq_zero_kernel<<<(WS_FLOATS + 255) / 256, 256, 0, stream>>>(ws, WS_FLOATS);

    vq_main_kernel<<<N_TOKENS / TOK_PER_BLOCK, 256, 0, stream>>>(
        z, codebook, out + OQ, out + OI, ws_counts, ws_dw, ws_loss);

    vq_finalize_kernel<<<1, 1024, 0, stream>>>(
        cluster_size, ema_w, ws_counts, ws_dw, ws_loss,
        out + OL, out + OCB, out + OCS, out + OEW);
}